// GraphEmbedder_8083128451311
// MI455X (gfx1250) — compile-verified
//
#include <hip/hip_runtime.h>

// ---------------- problem constants ----------------
#define N_NODES 10000
#define N_EDGES 160000
#define EMB     10
#define EPAD    16          // padded embedding width for 16x16 WMMA tiles
#define NT      625         // 16-row output tiles (625*16 == 10000)
#define KSPLIT  10          // K-dimension split for parallelism
#define KCHUNK  1000        // N_NODES / KSPLIT, multiple of 4

typedef __attribute__((ext_vector_type(2))) float v2f;
typedef __attribute__((ext_vector_type(4))) float v4f;
typedef __attribute__((ext_vector_type(8))) float v8f;

// ---------------- d_ws float-offset layout ----------------
#define WS_BIAS  0                        // 16 floats: ws_b + v_weights + wnp_b
#define WS_U     16                       // 16 floats: folded q-head vector u
#define WS_CONST 32                       // 1 float : folded q-head constant
#define WS_COLP  64                       // 40*16 floats: per-block column partials
#define WS_EA    1024                     // 160000 floats: E buffer A (interleaved padded)
#define WS_EB    (WS_EA + N_NODES*EPAD)   // 160000 floats: E buffer B
#define WS_PART  (WS_EB + N_NODES*EPAD)   // KSPLIT*NT*256 floats: WMMA partial tiles

// Interleaved padded E layout: E[i][j] at (i>>1)*32 + j*2 + (i&1).
// This makes the WMMA B operand (lane holds E[k+2h][col], E[k+2h+1][col])
// a single 8-byte contiguous load per lane.
__device__ __forceinline__ int eidx(int i, int j) { return ((i >> 1) * 32) + (j * 2) + (i & 1); }

// ---------------- 1) edge-embedding reduction -> biasTot ----------------
__global__ __launch_bounds__(1024)
void k_edge(const float* __restrict__ wts, const float* __restrict__ wew_w,
            const float* __restrict__ wew_b, const float* __restrict__ wnw_w,
            const float* __restrict__ wnw_b, const float* __restrict__ ws_b,
            const float* __restrict__ wnp_b, float* __restrict__ ws) {
  __shared__ float red[1024];
  __shared__ float ssum[EMB];
  const int tid = threadIdx.x;
  float acc[EMB], w0[EMB], b0[EMB];
  for (int j = 0; j < EMB; ++j) { acc[j] = 0.f; w0[j] = wew_w[j]; b0[j] = wew_b[j]; }
  for (int e = tid; e < N_EDGES; e += 1024) {
    const float w = wts[e];
    for (int j = 0; j < EMB; ++j) {
      float v = fmaf(w, w0[j], b0[j]);
      acc[j] += (v > 0.f) ? v : 0.f;
    }
  }
  for (int j = 0; j < EMB; ++j) {           // fixed-order tree reduction (deterministic)
    red[tid] = acc[j]; __syncthreads();
    for (int s = 512; s > 0; s >>= 1) { if (tid < s) red[tid] += red[tid + s]; __syncthreads(); }
    if (tid == 0) ssum[j] = red[0];
    __syncthreads();
  }
  if (tid < EMB) {
    float v = wnw_b[tid];
    for (int jp = 0; jp < EMB; ++jp) v = fmaf(ssum[jp], wnw_w[jp * EMB + tid], v);
    ws[WS_BIAS + tid] = v + ws_b[tid] + wnp_b[tid];   // v_weights + ws_b + wnp_b
  }
}

// ---------------- 2) iteration-1 shortcut: E0 = relu(f*ws_w + biasTot) ----------------
__global__ __launch_bounds__(256)
void k_initE(const float* __restrict__ feat, const float* __restrict__ ws_w,
             const float* __restrict__ ws, float* __restrict__ Eout) {
  const int i = blockIdx.x * blockDim.x + threadIdx.x;
  if (i >= N_NODES) return;
  const float f = feat[i];
  for (int j = 0; j < EPAD; ++j) {
    float v = 0.f;
    if (j < EMB) { v = fmaf(f, ws_w[j], ws[WS_BIAS + j]); v = (v > 0.f) ? v : 0.f; }
    Eout[eidx(i, j)] = v;
  }
}

// ---------------- 3) streaming WMMA: partial[kc][tile] = A[tile, kc-chunk] @ E ----------------
__global__ __launch_bounds__(128)
void k_wmma(const float* __restrict__ adj, const float* __restrict__ Ein,
            float* __restrict__ part) {
  const int wave = threadIdx.x >> 5;
  const int lane = threadIdx.x & 31;
  const int tile = blockIdx.x * 4 + wave;
  if (tile >= NT) return;                    // whole-wave early exit; active waves keep EXEC=~0
  const int kc  = blockIdx.y;
  const int h   = lane >> 4;                 // lane-half selects K pair (K, K+1) vs (K+2, K+3)
  const int col = lane & 15;                 // A: row M within tile; B/C/D: column N
  const int row = tile * 16 + col;

  const float* pA = adj + (size_t)row * N_NODES + (size_t)(kc * KCHUNK + 2 * h);
  const float* pB = Ein + (size_t)((kc * KCHUNK) >> 1) * 32 + (size_t)h * 32 + col * 2;

  v8f acc = {0.f, 0.f, 0.f, 0.f, 0.f, 0.f, 0.f, 0.f};
#pragma unroll 4
  for (int k = 0; k < KCHUNK; k += 4) {
    const v2f a = *(const v2f*)pA;           // A[M=col][K+2h], A[M=col][K+2h+1]
    const v2f b = *(const v2f*)pB;           // E[K+2h][col],   E[K+2h+1][col]
    acc = __builtin_amdgcn_wmma_f32_16x16x4_f32(
        /*neg_a=*/false, a, /*neg_b=*/false, b,
        /*c_mod=*/(short)0, acc, /*reuse_a=*/false, /*reuse_b=*/false);
    pA += 4;
    pB += 64;
  }
  // C/D layout: VGPR v -> row v (lanes 0-15) / row v+8 (lanes 16-31), col = lane&15
  float* pc = part + ((size_t)kc * NT + tile) * 256 + (size_t)(8 * h) * 16 + col;
#pragma unroll
  for (int v = 0; v < 8; ++v) pc[v * 16] = acc[v];
}

// ---------------- 4) reduce K-splits, apply wnp + bias + relu -> next E ----------------
__global__ __launch_bounds__(256)
void k_reduce(const float* __restrict__ part, const float* __restrict__ feat,
              const float* __restrict__ ws_w, const float* __restrict__ wnp_w,
              const float* __restrict__ ws, float* __restrict__ Eout,
              float* __restrict__ emb_out, int finalpass) {
  const int i = blockIdx.x * blockDim.x + threadIdx.x;
  if (i >= N_NODES) return;
  const int t = i >> 4, r = i & 15;
  float S[EPAD];
  for (int j = 0; j < EPAD; ++j) S[j] = 0.f;
  for (int kc = 0; kc < KSPLIT; ++kc) {      // fixed kc order -> deterministic
    const float* p = part + ((size_t)kc * NT + t) * 256 + r * 16;
    for (int j4 = 0; j4 < 4; ++j4) {
      const v4f v = *(const v4f*)(p + j4 * 4);
      S[j4 * 4 + 0] += v[0]; S[j4 * 4 + 1] += v[1];
      S[j4 * 4 + 2] += v[2]; S[j4 * 4 + 3] += v[3];
    }
  }
  const float f = feat[i];
  for (int j = 0; j < EMB; ++j) {
    float v = ws[WS_BIAS + j];
    v = fmaf(f, ws_w[j], v);
    for (int jp = 0; jp < EMB; ++jp) v = fmaf(S[jp], wnp_w[jp * EMB + j], v);
    v = (v > 0.f) ? v : 0.f;
    Eout[eidx(i, j)] = v;
    if (finalpass) emb_out[i * EMB + j] = v;
  }
  for (int j = EMB; j < EPAD; ++j) Eout[eidx(i, j)] = 0.f;
}

// ---------------- 5) deterministic column sums of final E ----------------
__global__ __launch_bounds__(256)
void k_colsum(const float* __restrict__ Ein, float* __restrict__ colp) {
  __shared__ float red[256];
  const int tid = threadIdx.x;
  const int i = blockIdx.x * 256 + tid;
  float acc[EMB];
  if (i < N_NODES) { for (int j = 0; j < EMB; ++j) acc[j] = Ein[eidx(i, j)]; }
  else             { for (int j = 0; j < EMB; ++j) acc[j] = 0.f; }
  for (int j = 0; j < EMB; ++j) {
    red[tid] = acc[j]; __syncthreads();
    for (int s = 128; s > 0; s >>= 1) { if (tid < s) red[tid] += red[tid + s]; __syncthreads(); }
    if (tid == 0) colp[blockIdx.x * 16 + j] = red[0];
    __syncthreads();
  }
}

// ---------------- 6) fold q-head: q[i] = const2 + E2[i] . u ----------------
__global__ void k_final_small(const float* __restrict__ colp, const float* __restrict__ wqa_w,
                              const float* __restrict__ wqa_b, const float* __restrict__ wqr_w,
                              const float* __restrict__ wqr_b, const float* __restrict__ wqact_w,
                              const float* __restrict__ wqact_b, float* __restrict__ ws) {
  if (threadIdx.x != 0) return;              // tiny, serial, deterministic
  float cs[EMB], se[EMB];
  for (int j = 0; j < EMB; ++j) {
    float s = 0.f;
    for (int b = 0; b < 40; ++b) s += colp[b * 16 + j];
    cs[j] = s;
  }
  for (int j = 0; j < EMB; ++j) {
    float v = wqa_b[j];
    for (int jp = 0; jp < EMB; ++jp) v = fmaf(cs[jp], wqa_w[jp * EMB + j], v);
    se[j] = v;                               // sumembed
  }
  float c = wqr_b[0];
  for (int j = 0; j < EMB; ++j) c = fmaf(se[j], wqr_w[j], c);
  for (int j = 0; j < EMB; ++j) c = fmaf(wqact_b[j], wqr_w[EMB + j], c);
  for (int jp = 0; jp < EMB; ++jp) {
    float u = 0.f;
    for (int j = 0; j < EMB; ++j) u = fmaf(wqact_w[jp * EMB + j], wqr_w[EMB + j], u);
    ws[WS_U + jp] = u;
  }
  ws[WS_CONST] = c;
}

__global__ __launch_bounds__(256)
void k_q(const float* __restrict__ Ein, const float* __restrict__ ws, float* __restrict__ q) {
  const int i = blockIdx.x * blockDim.x + threadIdx.x;
  if (i >= N_NODES) return;
  float v = ws[WS_CONST];
  for (int j = 0; j < EMB; ++j) v = fmaf(Ein[eidx(i, j)], ws[WS_U + j], v);
  q[i] = v;
}

// ---------------- launcher ----------------
extern "C" void kernel_launch(void* const* d_in, const int* in_sizes, int n_in,
                              void* d_out, int out_size, void* d_ws, size_t ws_size,
                              hipStream_t stream) {
  const float* feat    = (const float*)d_in[0];
  const float* wts     = (const float*)d_in[1];
  const float* adj     = (const float*)d_in[2];
  const float* ws_w    = (const float*)d_in[3];
  const float* ws_b    = (const float*)d_in[4];
  const float* wnp_w   = (const float*)d_in[5];
  const float* wnp_b   = (const float*)d_in[6];
  const float* wnw_w   = (const float*)d_in[7];
  const float* wnw_b   = (const float*)d_in[8];
  const float* wew_w   = (const float*)d_in[9];
  const float* wew_b   = (const float*)d_in[10];
  const float* wqr_w   = (const float*)d_in[11];
  const float* wqr_b   = (const float*)d_in[12];
  const float* wqa_w   = (const float*)d_in[13];
  const float* wqa_b   = (const float*)d_in[14];
  const float* wqact_w = (const float*)d_in[15];
  const float* wqact_b = (const float*)d_in[16];

  float* out  = (float*)d_out;               // [0..N): q_vals, [N..N+N*EMB): embeddings
  float* ws   = (float*)d_ws;
  float* EA   = ws + WS_EA;
  float* EB   = ws + WS_EB;
  float* part = ws + WS_PART;

  const dim3 gmm((NT + 3) / 4, KSPLIT);      // 4 waves/block, 16-row tile per wave

  k_edge<<<1, 1024, 0, stream>>>(wts, wew_w, wew_b, wnw_w, wnw_b, ws_b, wnp_b, ws);
  k_initE<<<40, 256, 0, stream>>>(feat, ws_w, ws, EA);                 // iter 1 (A@0 == 0)
  k_wmma <<<gmm, 128, 0, stream>>>(adj, EA, part);                     // iter 2 matmul
  k_reduce<<<40, 256, 0, stream>>>(part, feat, ws_w, wnp_w, ws, EB, out + N_NODES, 0);
  k_wmma <<<gmm, 128, 0, stream>>>(adj, EB, part);                     // iter 3 matmul
  k_reduce<<<40, 256, 0, stream>>>(part, feat, ws_w, wnp_w, ws, EA, out + N_NODES, 1);
  k_colsum<<<40, 256, 0, stream>>>(EA, ws + WS_COLP);
  k_final_small<<<1, 32, 0, stream>>>(ws + WS_COLP, wqa_w, wqa_b, wqr_w, wqr_b,
                                      wqact_w, wqact_b, ws);
  k_q<<<40, 256, 0, stream>>>(EA, ws, out);
}